// ArvaneModel_59459527246075
// MI455X (gfx1250) — compile-verified
//
#include <hip/hip_runtime.h>
#include <math.h>

// Reference dims (compile-time constants)
#define V    8
#define NC   47
#define FH   60
#define FW   80
#define DH   240
#define DW   320
#define NPTS 150000

typedef __attribute__((ext_vector_type(2))) float v2f;
typedef __attribute__((ext_vector_type(8))) float v8f;

__global__ __launch_bounds__(256) void ArvaneModel_fused_kernel(
    const float* __restrict__ feats,   // (V, NC, FH, FW)
    const float* __restrict__ depthm,  // (V, DH, DW)
    const float* __restrict__ poses,   // (V, 4, 4)
    const float* __restrict__ Kc,      // (V, 3, 3)
    const float* __restrict__ Kd,      // (V, 3, 3)
    const float* __restrict__ coords,  // (NPTS, 3)
    float* __restrict__ out)           // density (NC,NPTS) ++ wsum (NPTS) ++ count (NPTS)
{
    // Per-block projection table: per view, 5 rows x 4 cols:
    // row0 = Kc[0,:]·inv(pose)[0:3,:], row1 = Kc[1,:]·..., row2 = inv(pose)[2,:],
    // row3 = Kd[0,:]·..., row4 = Kd[1,:]·...
    __shared__ __align__(16) float sP[V * 20];

    const int tid = threadIdx.x;
    if (tid < V) {
        const int v = tid;
        float m[4][4], inv[4][4];
        for (int i = 0; i < 4; ++i)
            for (int k = 0; k < 4; ++k) {
                m[i][k]   = poses[v * 16 + i * 4 + k];
                inv[i][k] = (i == k) ? 1.0f : 0.0f;
            }
        // Gauss-Jordan with partial pivoting (4x4)
        for (int col = 0; col < 4; ++col) {
            int piv = col;
            float best = fabsf(m[col][col]);
            for (int r = col + 1; r < 4; ++r) {
                float t = fabsf(m[r][col]);
                if (t > best) { best = t; piv = r; }
            }
            if (piv != col) {
                for (int k = 0; k < 4; ++k) {
                    float t = m[col][k]; m[col][k] = m[piv][k]; m[piv][k] = t;
                    t = inv[col][k]; inv[col][k] = inv[piv][k]; inv[piv][k] = t;
                }
            }
            float s = 1.0f / m[col][col];
            for (int k = 0; k < 4; ++k) { m[col][k] *= s; inv[col][k] *= s; }
            for (int r = 0; r < 4; ++r) {
                if (r == col) continue;
                float f2 = m[r][col];
                for (int k = 0; k < 4; ++k) {
                    m[r][k]   -= f2 * m[col][k];
                    inv[r][k] -= f2 * inv[col][k];
                }
            }
        }
        for (int jj = 0; jj < 4; ++jj) {
            float r0 = inv[0][jj], r1 = inv[1][jj], r2 = inv[2][jj];
            sP[v * 20 + 0 * 4 + jj] = Kc[v * 9 + 0] * r0 + Kc[v * 9 + 1] * r1 + Kc[v * 9 + 2] * r2;
            sP[v * 20 + 1 * 4 + jj] = Kc[v * 9 + 3] * r0 + Kc[v * 9 + 4] * r1 + Kc[v * 9 + 5] * r2;
            sP[v * 20 + 2 * 4 + jj] = r2;
            sP[v * 20 + 3 * 4 + jj] = Kd[v * 9 + 0] * r0 + Kd[v * 9 + 1] * r1 + Kd[v * 9 + 2] * r2;
            sP[v * 20 + 4 * 4 + jj] = Kd[v * 9 + 3] * r0 + Kd[v * 9 + 4] * r1 + Kd[v * 9 + 5] * r2;
        }
    }
    __syncthreads();

    // ---- Wave/WMMA geometry: each wave32 owns 32 consecutive points ----
    const int lane  = tid & 31;
    const int half  = lane >> 4;        // 0: lanes 0-15 (K=0,1), 1: lanes 16-31 (K=2,3)
    const int j     = lane & 15;
    const int wbase = blockIdx.x * 256 + (tid & ~31);
    const int n     = wbase + lane;
    const bool active = (n < NPTS);

    // B matrices: xyz1 columns for points [wbase .. wbase+15] and [wbase+16 .. wbase+31]
    const int q1 = min(wbase + j,       NPTS - 1);
    const int q2 = min(wbase + 16 + j,  NPTS - 1);
    const float x1c = coords[q1 * 3 + 0], y1c = coords[q1 * 3 + 1], z1c = coords[q1 * 3 + 2];
    const float x2c = coords[q2 * 3 + 0], y2c = coords[q2 * 3 + 1], z2c = coords[q2 * 3 + 2];
    v2f B1, B2;
    B1.x = half ? z1c : x1c;  B1.y = half ? 1.0f : y1c;   // K-rows {0,1} vs {2,3}
    B2.x = half ? z2c : x2c;  B2.y = half ? 1.0f : y2c;

    const int  k0   = half ? 2 : 0;
    const bool inA1 = (j < 5);               // A-tile 1: rows M=0..4 hold P rows 0..4
    const bool inA2 = (j >= 8) && (j < 13);  // A-tile 2: rows M=8..12 hold P rows 0..4
    const int  r1   = inA1 ? j : 0;          // clamped -> LDS read always in-bounds
    const int  r2   = inA2 ? (j - 8) : 0;

    int   bofs[V];
    float w00[V], w01[V], w10[V], w11[V];
    float wsum = 0.0f, cnt = 0.0f;

#pragma unroll
    for (int v = 0; v < V; ++v) {
        // Unconditional 8B-aligned LDS loads (dword offset is always even),
        // then branch-free select -> ds_load_b64 + v_cndmask, no exec masking.
        const v2f p1 = *(const v2f*)&sP[v * 20 + r1 * 4 + k0];
        const v2f p2 = *(const v2f*)&sP[v * 20 + r2 * 4 + k0];
        v2f A1, A2;
        A1.x = inA1 ? p1.x : 0.0f;  A1.y = inA1 ? p1.y : 0.0f;
        A2.x = inA2 ? p2.x : 0.0f;  A2.y = inA2 ? p2.y : 0.0f;

        v8f d = {};
        // D = A1(16x4) x B1(4x16): lanes 0-15 (rows 0-7) get points wbase+0..15
        d = __builtin_amdgcn_wmma_f32_16x16x4_f32(false, A1, false, B1, (short)0, d, false, false);
        // accumulate A2 (rows 8-12) x B2: lanes 16-31 (rows 8-15) get points wbase+16..31
        d = __builtin_amdgcn_wmma_f32_16x16x4_f32(false, A2, false, B2, (short)0, d, false, false);

        // Every lane now holds ITS point's projection in d[0..4]
        const float cxc = d[0], cyc = d[1], cz = d[2], cxd = d[3], cyd = d[4];

        const float zs   = (fabsf(cz) > 1e-6f) ? cz : 1e-6f;
        const float invz = 1.0f / zs;          // one IEEE div, four muls
        const float uc = cxc * invz, vc = cyc * invz;
        const float ud = cxd * invz, vd = cyd * invz;

        // bilinear sample setup on color grid
        const bool validc = (uc >= 0.0f) && (uc <= (float)(FW - 1)) &&
                            (vc >= 0.0f) && (vc <= (float)(FH - 1));
        const float ucl = fminf(fmaxf(uc, 0.0f), (float)(FW - 1));
        const float vcl = fminf(fmaxf(vc, 0.0f), (float)(FH - 1));
        const int x0 = min((int)floorf(ucl), FW - 2);
        const int y0 = min((int)floorf(vcl), FH - 2);
        const float wx = ucl - (float)x0;   // in [0,1]; ==1 at right/bottom edge (same math)
        const float wy = vcl - (float)y0;

        // nearest sample on depth grid (jnp.round == half-to-even == rintf)
        const bool validd = (ud >= 0.0f) && (ud <= (float)(DW - 1)) &&
                            (vd >= 0.0f) && (vd <= (float)(DH - 1));
        const float udc = fminf(fmaxf(ud, 0.0f), (float)(DW - 1));
        const float vdc = fminf(fmaxf(vd, 0.0f), (float)(DH - 1));
        const int xi = (int)rintf(udc);
        const int yi = (int)rintf(vdc);
        const float dsmp = depthm[v * (DH * DW) + yi * DW + xi];

        const bool  valid = validc && validd && (cz > 0.0f) && (dsmp > 0.0f);
        const float dist  = (cz - dsmp) * 16.0f;
        const float w     = valid ? __expf(-dist * dist) : 0.0f;

        wsum += w;
        cnt  += valid ? 1.0f : 0.0f;

        bofs[v] = v * (NC * FH * FW) + y0 * FW + x0;
        const float omx = 1.0f - wx, omy = 1.0f - wy;
        w00[v] = omy * omx * w;
        w01[v] = omy * wx  * w;
        w10[v] = wy  * omx * w;
        w11[v] = wy  * wx  * w;
    }

    // ---- channel loop: 8 views x fixed 2x2 stencil, weight pre-folded ----
    for (int c = 0; c < NC; ++c) {
        float acc = 0.0f;
        const int choff = c * (FH * FW);
#pragma unroll
        for (int v = 0; v < V; ++v) {
            const float* f = feats + bofs[v] + choff;
            acc += f[0]      * w00[v] + f[1]      * w01[v]
                 + f[FW]     * w10[v] + f[FW + 1] * w11[v];
        }
        if (active) __builtin_nontemporal_store(acc, &out[c * NPTS + n]);
    }
    if (active) {
        __builtin_nontemporal_store(wsum, &out[NC * NPTS + n]);
        __builtin_nontemporal_store(cnt,  &out[(NC + 1) * NPTS + n]);
    }
}

extern "C" void kernel_launch(void* const* d_in, const int* in_sizes, int n_in,
                              void* d_out, int out_size, void* d_ws, size_t ws_size,
                              hipStream_t stream) {
    const float* feats  = (const float*)d_in[0];
    const float* depthm = (const float*)d_in[1];
    const float* poses  = (const float*)d_in[2];
    const float* Kc     = (const float*)d_in[3];
    const float* Kd     = (const float*)d_in[4];
    const float* coords = (const float*)d_in[5];
    float* out = (float*)d_out;

    const int threads = 256;
    const int blocks  = (NPTS + threads - 1) / threads;
    hipLaunchKernelGGL(ArvaneModel_fused_kernel, dim3(blocks), dim3(threads), 0, stream,
                       feats, depthm, poses, Kc, Kd, coords, out);
}